// PagedOffloadTransformer_67173288509653
// MI455X (gfx1250) — compile-verified
//
#include <hip/hip_runtime.h>
#include <hip/hip_bf16.h>

#define B_ 8
#define L_ 2
#define DIM_ 1024
#define H_ 8
#define HD_ 128
#define BS_ 16
#define SEQ_ 2048
#define NB_ 129
#define SCALE_ (0.08838834764831845f)   /* 1/sqrt(128) */
#define EPS_ 1e-5f
#define SPLITK_ 8

typedef __attribute__((ext_vector_type(2))) float v2f;
typedef __attribute__((ext_vector_type(8))) float v8f;

// ---------------------------------------------------------------------------
// RMSNorm: one block per row (B rows), 256 threads, DIM=1024
// ---------------------------------------------------------------------------
__global__ void rmsnorm_kernel(const float* __restrict__ x,
                               const float* __restrict__ w,
                               float* __restrict__ out) {
    __shared__ float red[256];
    const int row = blockIdx.x;
    const float* xr = x + row * DIM_;
    float s = 0.f;
#pragma unroll
    for (int i = 0; i < DIM_ / 256; ++i) {
        float v = xr[threadIdx.x + i * 256];
        s += v * v;
    }
    red[threadIdx.x] = s;
    __syncthreads();
    for (int off = 128; off > 0; off >>= 1) {
        if (threadIdx.x < off) red[threadIdx.x] += red[threadIdx.x + off];
        __syncthreads();
    }
    const float r = rsqrtf(red[0] / (float)DIM_ + EPS_);
    float* orow = out + row * DIM_;
#pragma unroll
    for (int i = 0; i < DIM_ / 256; ++i) {
        int c = threadIdx.x + i * 256;
        orow[c] = xr[c] * r * w[c];
    }
}

// ---------------------------------------------------------------------------
// Split-K multi-GEMM: partial[g][s][8][N] = A[16,K] @ Wg[N,K]^T over k-chunk s.
// A has 16 rows in memory with rows 8..15 pre-zeroed => unconditional loads.
// WMMA f32 16x16x4; one wave per (g, k-chunk, 16-wide N tile); 4 acc chains.
// A layout (16x4): lane&15 = M row, lane>>4 selects K pair {0,1}/{2,3}.
// B layout (4x16): lane&15 = N col, lane>>4 selects K pair (mirror of A).
// C/D layout (16x16): VGPR j -> row j (lanes 0-15) / row j+8 (lanes 16-31).
// ---------------------------------------------------------------------------
__global__ void gemm_wmma_splitk(const float* __restrict__ A,
                                 const float* __restrict__ W0,
                                 const float* __restrict__ W1,
                                 const float* __restrict__ W2,
                                 float* __restrict__ partial,
                                 int N, int K, int nW) {
    const int wave = threadIdx.x >> 5;
    const int lane = threadIdx.x & 31;
    const int t = blockIdx.x * (blockDim.x >> 5) + wave;
    const int tilesPerW = (N >> 4) * SPLITK_;
    if (t >= nW * tilesPerW) return;

    const int g   = t / tilesPerW;
    const int rem = t - g * tilesPerW;
    const int s   = rem / (N >> 4);
    const int nt  = rem - s * (N >> 4);
    const int n0  = nt << 4;

    const float* W = (g == 0) ? W0 : ((g == 1) ? W1 : W2);
    const int kc  = K / SPLITK_;
    const int k0  = s * kc;
    const int kh2 = (lane >> 4) << 1;   // K sub-offset for this half-wave
    const int l16 = lane & 15;

    const float* ap = A + (size_t)l16 * (size_t)K + k0 + kh2;
    const float* wp = W + (size_t)(n0 + l16) * (size_t)K + k0 + kh2;

    v8f c0 = {0.f, 0.f, 0.f, 0.f, 0.f, 0.f, 0.f, 0.f};
    v8f c1 = {0.f, 0.f, 0.f, 0.f, 0.f, 0.f, 0.f, 0.f};
    v8f c2 = {0.f, 0.f, 0.f, 0.f, 0.f, 0.f, 0.f, 0.f};
    v8f c3 = {0.f, 0.f, 0.f, 0.f, 0.f, 0.f, 0.f, 0.f};

    for (int k = 0; k < kc; k += 16) {
        v2f a0 = *(const v2f*)(ap);
        v2f b0 = __builtin_nontemporal_load((const v2f*)(wp));
        v2f a1 = *(const v2f*)(ap + 4);
        v2f b1 = __builtin_nontemporal_load((const v2f*)(wp + 4));
        v2f a2 = *(const v2f*)(ap + 8);
        v2f b2 = __builtin_nontemporal_load((const v2f*)(wp + 8));
        v2f a3 = *(const v2f*)(ap + 12);
        v2f b3 = __builtin_nontemporal_load((const v2f*)(wp + 12));
        c0 = __builtin_amdgcn_wmma_f32_16x16x4_f32(false, a0, false, b0,
                                                   (short)0, c0, false, false);
        c1 = __builtin_amdgcn_wmma_f32_16x16x4_f32(false, a1, false, b1,
                                                   (short)0, c1, false, false);
        c2 = __builtin_amdgcn_wmma_f32_16x16x4_f32(false, a2, false, b2,
                                                   (short)0, c2, false, false);
        c3 = __builtin_amdgcn_wmma_f32_16x16x4_f32(false, a3, false, b3,
                                                   (short)0, c3, false, false);
        ap += 16;
        wp += 16;
    }

    if ((lane >> 4) == 0) {             // rows 0..7 live in lanes 0-15
        float* pout = partial + (size_t)(g * SPLITK_ + s) * 8 * (size_t)N
                              + (size_t)(n0 + l16);
#pragma unroll
        for (int j = 0; j < 8; ++j)
            pout[(size_t)j * (size_t)N] = c0[j] + c1[j] + c2[j] + c3[j];
    }
}

// ---------------------------------------------------------------------------
// Split-K reduction: out[i] = sum_s partial[s][i] (+ residual[i])
// ---------------------------------------------------------------------------
__global__ void splitk_reduce_kernel(const float* __restrict__ partial,
                                     const float* __restrict__ residual,
                                     float* __restrict__ out, int n) {
    int i = blockIdx.x * blockDim.x + threadIdx.x;
    if (i < n) {
        float s = 0.f;
#pragma unroll
        for (int j = 0; j < SPLITK_; ++j) s += partial[(size_t)j * n + i];
        if (residual) s += residual[i];
        out[i] = s;
    }
}

// ---------------------------------------------------------------------------
// SiLU(g) * u, elementwise
// ---------------------------------------------------------------------------
__global__ void silu_mul_kernel(const float* __restrict__ g,
                                const float* __restrict__ u,
                                float* __restrict__ e, int n) {
    int i = blockIdx.x * blockDim.x + threadIdx.x;
    if (i < n) {
        float gv = g[i];
        e[i] = gv * (1.f / (1.f + __expf(-gv))) * u[i];
    }
}

// ---------------------------------------------------------------------------
// Paged flash-decode attention. One workgroup per (b,h); 8 waves split the
// 129 KV blocks; online softmax per wave; LDS combine. The new token's K/V
// are sourced from k_new/v_new at the slot given by slot_mapping (the input
// heaps are never mutated).
// ---------------------------------------------------------------------------
__global__ void paged_attn_kernel(const float* __restrict__ q,
                                  const float* __restrict__ k_new,
                                  const float* __restrict__ v_new,
                                  const float* __restrict__ key_heap,
                                  const float* __restrict__ val_heap,
                                  const int* __restrict__ bt,             // [B,NB] (layer slice)
                                  const long long* __restrict__ slot_map, // [B]   (layer slice)
                                  const int* __restrict__ ctx_lens,
                                  float* __restrict__ attn_out) {
    __shared__ float s_acc[8][HD_];
    __shared__ float s_m[8];
    __shared__ float s_l[8];

    const int b = blockIdx.x / H_;
    const int h = blockIdx.x % H_;
    const int wave = threadIdx.x >> 5;
    const int lane = threadIdx.x & 31;

    const int ctx = ctx_lens[b];
    const int nblocks = (ctx + BS_ - 1) / BS_;
    const long long slot = slot_map[b];

    const size_t qoff = ((size_t)b * H_ + h) * HD_;
    const float4 qv  = *(const float4*)(q + qoff + lane * 4);
    const float4 knv = *(const float4*)(k_new + qoff + lane * 4);
    const float4 vnv = *(const float4*)(v_new + qoff + lane * 4);

    float m = -1e30f, lsum = 0.f;
    float4 acc = {0.f, 0.f, 0.f, 0.f};

    for (int n = wave; n < nblocks; n += 8) {
        const int phys = bt[b * NB_ + n];
        const size_t blkbase = (size_t)phys * (size_t)(BS_ * H_ * HD_) + (size_t)h * HD_;
        for (int s = 0; s < BS_; ++s) {
            const int pos = n * BS_ + s;
            if (pos >= ctx) break;
            const long long sl = (long long)phys * BS_ + s;
            const bool is_new = (sl == slot);
            const size_t off = blkbase + (size_t)s * (H_ * HD_) + lane * 4;
            float4 kv = is_new ? knv : *(const float4*)(key_heap + off);
            float p = qv.x * kv.x + qv.y * kv.y + qv.z * kv.z + qv.w * kv.w;
#pragma unroll
            for (int d = 16; d > 0; d >>= 1) p += __shfl_xor(p, d, 32);
            p *= SCALE_;
            const float mnew = fmaxf(m, p);
            const float corr = __expf(m - mnew);
            const float wgt  = __expf(p - mnew);
            float4 vv = is_new ? vnv : *(const float4*)(val_heap + off);
            acc.x = acc.x * corr + wgt * vv.x;
            acc.y = acc.y * corr + wgt * vv.y;
            acc.z = acc.z * corr + wgt * vv.z;
            acc.w = acc.w * corr + wgt * vv.w;
            lsum = lsum * corr + wgt;
            m = mnew;
        }
    }

    s_acc[wave][lane * 4 + 0] = acc.x;
    s_acc[wave][lane * 4 + 1] = acc.y;
    s_acc[wave][lane * 4 + 2] = acc.z;
    s_acc[wave][lane * 4 + 3] = acc.w;
    if (lane == 0) { s_m[wave] = m; s_l[wave] = lsum; }
    __syncthreads();

    if (threadIdx.x < HD_) {
        const int d = threadIdx.x;
        float M = -1e30f;
#pragma unroll
        for (int w = 0; w < 8; ++w) M = fmaxf(M, s_m[w]);
        float num = 0.f, den = 0.f;
#pragma unroll
        for (int w = 0; w < 8; ++w) {
            const float f = __expf(s_m[w] - M);
            num += f * s_acc[w][d];
            den += f * s_l[w];
        }
        attn_out[qoff + d] = num / den;
    }
}

// ---------------------------------------------------------------------------
// Host orchestration
// ---------------------------------------------------------------------------
extern "C" void kernel_launch(void* const* d_in, const int* in_sizes, int n_in,
                              void* d_out, int out_size, void* d_ws, size_t ws_size,
                              hipStream_t stream) {
    (void)in_sizes; (void)n_in; (void)out_size; (void)ws_size;

    const float*     x            = (const float*)d_in[0];
    const float*     key_heap     = (const float*)d_in[1];
    const float*     val_heap     = (const float*)d_in[2];
    const int*       block_tables = (const int*)d_in[3];
    const long long* slot_mapping = (const long long*)d_in[4];
    const int*       context_lens = (const int*)d_in[5];
    const float*     wq = (const float*)d_in[6];
    const float*     wk = (const float*)d_in[7];
    const float*     wv = (const float*)d_in[8];
    const float*     wo = (const float*)d_in[9];
    const float*     w1 = (const float*)d_in[10];
    const float*     w2 = (const float*)d_in[11];
    const float*     w3 = (const float*)d_in[12];
    const float*     norm1_w = (const float*)d_in[13];
    const float*     norm2_w = (const float*)d_in[14];
    const float*     norm_w  = (const float*)d_in[15];

    float* ws      = (float*)d_ws;
    float* x_cur   = ws;                //  8192
    float* hbuf    = ws + 8192;         // 16384 (16 x 1024, rows 8-15 zero)
    float* attn    = ws + 24576;        // 16384 (16 x 1024, rows 8-15 zero)
    float* ebuf    = ws + 40960;        // 65536 (16 x 4096, rows 8-15 zero)
    float* qbuf    = ws + 106496;       //  8192
    float* kbuf    = ws + 114688;       //  8192
    float* vbuf    = ws + 122880;       //  8192
    float* gbuf    = ws + 131072;       // 32768
    float* ubuf    = ws + 163840;       // 32768
    float* partial = ws + 196608;       // 524288 (max: 2 * 8 * 8 * 4096)

    // rows 8..15 of every GEMM-A buffer must be zero (M padded 8 -> 16)
    hipMemsetAsync(hbuf + 8 * DIM_,     0, (size_t)8 * DIM_ * sizeof(float), stream);
    hipMemsetAsync(attn + 8 * DIM_,     0, (size_t)8 * DIM_ * sizeof(float), stream);
    hipMemsetAsync(ebuf + 8 * 4 * DIM_, 0, (size_t)8 * 4 * DIM_ * sizeof(float), stream);
    hipMemcpyAsync(x_cur, x, (size_t)B_ * DIM_ * sizeof(float),
                   hipMemcpyDeviceToDevice, stream);

    const size_t wsz = (size_t)DIM_ * DIM_;       // per-layer square weight
    const size_t fsz = (size_t)4 * DIM_ * DIM_;   // per-layer FFN weight

    for (int l = 0; l < L_; ++l) {
        // --- attention sub-block ---
        rmsnorm_kernel<<<B_, 256, 0, stream>>>(x_cur, norm1_w + l * DIM_, hbuf);

        // fused Q/K/V projections, split-K (3 * 64 tiles * 8 chunks = 1536 waves)
        gemm_wmma_splitk<<<384, 128, 0, stream>>>(hbuf, wq + l * wsz, wk + l * wsz,
                                                  wv + l * wsz, partial, DIM_, DIM_, 3);
        splitk_reduce_kernel<<<32, 256, 0, stream>>>(partial,               nullptr, qbuf, B_ * DIM_);
        splitk_reduce_kernel<<<32, 256, 0, stream>>>(partial + 8 * 8192,    nullptr, kbuf, B_ * DIM_);
        splitk_reduce_kernel<<<32, 256, 0, stream>>>(partial + 16 * 8192,   nullptr, vbuf, B_ * DIM_);

        paged_attn_kernel<<<B_ * H_, 256, 0, stream>>>(
            qbuf, kbuf, vbuf, key_heap, val_heap,
            block_tables + (size_t)l * B_ * NB_,
            slot_mapping + (size_t)l * B_,
            context_lens, attn);

        // output projection + residual
        gemm_wmma_splitk<<<128, 128, 0, stream>>>(attn, wo + l * wsz, wo + l * wsz,
                                                  wo + l * wsz, partial, DIM_, DIM_, 1);
        splitk_reduce_kernel<<<32, 256, 0, stream>>>(partial, x_cur, x_cur, B_ * DIM_);

        // --- SwiGLU FFN sub-block ---
        rmsnorm_kernel<<<B_, 256, 0, stream>>>(x_cur, norm2_w + l * DIM_, hbuf);

        // fused W1/W3 (2 * 256 tiles * 8 chunks = 4096 waves)
        gemm_wmma_splitk<<<1024, 128, 0, stream>>>(hbuf, w1 + l * fsz, w3 + l * fsz,
                                                   w3 + l * fsz, partial, 4 * DIM_, DIM_, 2);
        splitk_reduce_kernel<<<128, 256, 0, stream>>>(partial,              nullptr, gbuf, B_ * 4 * DIM_);
        splitk_reduce_kernel<<<128, 256, 0, stream>>>(partial + 8 * 32768,  nullptr, ubuf, B_ * 4 * DIM_);

        silu_mul_kernel<<<(B_ * 4 * DIM_ + 255) / 256, 256, 0, stream>>>(gbuf, ubuf, ebuf, B_ * 4 * DIM_);

        // down projection + residual (K = 4096)
        gemm_wmma_splitk<<<128, 128, 0, stream>>>(ebuf, w2 + l * fsz, w2 + l * fsz,
                                                  w2 + l * fsz, partial, DIM_, 4 * DIM_, 1);
        splitk_reduce_kernel<<<32, 256, 0, stream>>>(partial, x_cur, x_cur, B_ * DIM_);
    }

    rmsnorm_kernel<<<B_, 256, 0, stream>>>(x_cur, norm_w, (float*)d_out);
}